// BipartiteGraphConvolution_36842229465911
// MI455X (gfx1250) — compile-verified
//
#include <hip/hip_runtime.h>

// ---------------------------------------------------------------------------
// BipartiteGraphConvolution for MI455X (gfx1250), fp32 WMMA 16x16x4 path.
//
// Pipeline (all fp32):
//   K0 zero      : conv accumulator + BN stats scratch
//   K1 proj16    : left_proj = left @ W_left^T + b_left   (WMMA)
//                  right_proj = right @ W_right^T          (WMMA)
//   K2 edge_stats: column sum/sumsq of joint (recomputed, not stored)
//   K3 bn_final  : BN1 affine  a1[c] = g*rsqrt(var+eps), b1[c] = beta - mean*a1
//   K4 edge_gemm : A = relu(BN1(joint)) recomputed on the fly,
//                  D = A @ W_final^T + b_final (WMMA), scatter-add into conv
//   K5 colstats  : column sum/sumsq of conv
//   K6 bn_final  : BN2 affine
//   K7 output    : h = relu([BN2(conv), right] @ W_out1^T + b_out1)  (WMMA)
//                  out = relu(h @ W_out2^T + b_out2)                 (WMMA)
// ---------------------------------------------------------------------------

typedef float v2f __attribute__((ext_vector_type(2)));
typedef float v8f __attribute__((ext_vector_type(8)));

#define NL 50000
#define NR 100000
#define NE 1000000

static __device__ __forceinline__ v8f wmma_f32(v2f a, v2f b, v8f c) {
  // 8 args: (neg_a, A, neg_b, B, c_mod, C, reuse_a, reuse_b)
  return __builtin_amdgcn_wmma_f32_16x16x4_f32(false, a, false, b, (short)0, c,
                                               false, false);
}

// ---------------------------------------------------------------------------
__global__ void zero_f32(float* __restrict__ p, int n) {
  int i = blockIdx.x * blockDim.x + threadIdx.x;
  int stride = gridDim.x * blockDim.x;
  for (; i < n; i += stride) p[i] = 0.0f;
}

// ---------------------------------------------------------------------------
// Y[m][n] = sum_k X[m][k] * W[n][k] + bias[n]   (X: rows x 64, W: 64x64)
// One wave per 16-row tile, full N=64 (four 16x16 tiles), K=64 in steps of 4.
__global__ __launch_bounds__(32) void proj16(const float* __restrict__ X,
                                             const float* __restrict__ W,
                                             const float* __restrict__ bias,
                                             float* __restrict__ Y) {
  const int lane = threadIdx.x;
  const int mlane = lane & 15;
  const int khalf = lane >> 4;
  const int m0 = blockIdx.x * 16;

  v8f acc[4];
#pragma unroll
  for (int nt = 0; nt < 4; ++nt) {
    float bv = bias ? bias[nt * 16 + mlane] : 0.0f;
#pragma unroll
    for (int i = 0; i < 8; ++i) acc[nt][i] = bv;
  }

  const float* arow = X + (size_t)(m0 + mlane) * 64 + khalf * 2;
#pragma unroll
  for (int k0 = 0; k0 < 64; k0 += 4) {
    v2f a = *(const v2f*)(arow + k0);
#pragma unroll
    for (int nt = 0; nt < 4; ++nt) {
      v2f b = *(const v2f*)(W + (size_t)(nt * 16 + mlane) * 64 + k0 + khalf * 2);
      acc[nt] = wmma_f32(a, b, acc[nt]);
    }
  }

#pragma unroll
  for (int nt = 0; nt < 4; ++nt)
#pragma unroll
    for (int i = 0; i < 8; ++i)
      Y[(size_t)(m0 + i + khalf * 8) * 64 + nt * 16 + mlane] = acc[nt][i];
}

// ---------------------------------------------------------------------------
// Column sum / sumsq of joint (recomputed, never stored):
//   joint[e][c] = ef[e]*We[c] + lp[li[e]][c] + rp[ri[e]][c]
__global__ __launch_bounds__(256) void edge_stats(
    const int* __restrict__ li, const int* __restrict__ ri,
    const float* __restrict__ ef, const float* __restrict__ lp,
    const float* __restrict__ rp, const float* __restrict__ We,
    float* __restrict__ sum, float* __restrict__ sumsq, int ne) {
  const int c = threadIdx.x & 63;
  const int sub = threadIdx.x >> 6;  // 0..3
  const float w = We[c];
  float s = 0.0f, q = 0.0f;
  for (int e = blockIdx.x * 4 + sub; e < ne; e += gridDim.x * 4) {
    int l = li[e], r = ri[e];
    float v = ef[e] * w + lp[(size_t)l * 64 + c] + rp[(size_t)r * 64 + c];
    s += v;
    q += v * v;
  }
  __shared__ float rs[256], rq[256];
  rs[threadIdx.x] = s;
  rq[threadIdx.x] = q;
  __syncthreads();
  if (sub == 0) {
    atomicAdd(&sum[c], rs[c] + rs[c + 64] + rs[c + 128] + rs[c + 192]);
    atomicAdd(&sumsq[c], rq[c] + rq[c + 64] + rq[c + 128] + rq[c + 192]);
  }
}

// Column sum / sumsq of a dense row-major [nrows,64] matrix.
__global__ __launch_bounds__(256) void colstats(const float* __restrict__ X,
                                                int nrows,
                                                float* __restrict__ sum,
                                                float* __restrict__ sumsq) {
  const int c = threadIdx.x & 63;
  const int sub = threadIdx.x >> 6;
  float s = 0.0f, q = 0.0f;
  for (int r = blockIdx.x * 4 + sub; r < nrows; r += gridDim.x * 4) {
    float v = X[(size_t)r * 64 + c];
    s += v;
    q += v * v;
  }
  __shared__ float rs[256], rq[256];
  rs[threadIdx.x] = s;
  rq[threadIdx.x] = q;
  __syncthreads();
  if (sub == 0) {
    atomicAdd(&sum[c], rs[c] + rs[c + 64] + rs[c + 128] + rs[c + 192]);
    atomicAdd(&sumsq[c], rq[c] + rq[c + 64] + rq[c + 128] + rq[c + 192]);
  }
}

// BN affine: a[c] = gamma*rsqrt(var+eps), b[c] = beta - mean*a[c]
__global__ void bn_finalize(const float* __restrict__ sum,
                            const float* __restrict__ sumsq,
                            const float* __restrict__ gamma,
                            const float* __restrict__ beta, float n,
                            float* __restrict__ a, float* __restrict__ b) {
  int c = threadIdx.x;
  float mean = sum[c] / n;
  float var = sumsq[c] / n - mean * mean;
  float s = gamma[c] * rsqrtf(var + 1e-5f);
  a[c] = s;
  b[c] = beta[c] - mean * s;
}

// ---------------------------------------------------------------------------
// Per 16-edge tile: recompute joint, apply BN1 + relu while loading the A
// fragment, WMMA against W_final^T (+ b_final), then scatter-add D rows into
// conv[right_index] (this IS the segment_sum).
__global__ __launch_bounds__(32) void edge_gemm_scatter(
    const int* __restrict__ li, const int* __restrict__ ri,
    const float* __restrict__ ef, const float* __restrict__ lp,
    const float* __restrict__ rp, const float* __restrict__ We,
    const float* __restrict__ a1, const float* __restrict__ b1,
    const float* __restrict__ Wf, const float* __restrict__ bf,
    float* __restrict__ conv) {
  const int lane = threadIdx.x;
  const int mlane = lane & 15;
  const int khalf = lane >> 4;
  const int e0 = blockIdx.x * 16;

  __shared__ int sli[16], sri[16];
  __shared__ float sef[16];
  if (lane < 16) {
    sli[lane] = li[e0 + lane];
    sri[lane] = ri[e0 + lane];
    sef[lane] = ef[e0 + lane];
  }
  __syncthreads();

  const int l = sli[mlane];
  const int r = sri[mlane];
  const float ev = sef[mlane];
  const float* lrow = lp + (size_t)l * 64 + khalf * 2;
  const float* rrow = rp + (size_t)r * 64 + khalf * 2;

  v8f acc[4];
#pragma unroll
  for (int nt = 0; nt < 4; ++nt) {
    float bv = bf[nt * 16 + mlane];
#pragma unroll
    for (int i = 0; i < 8; ++i) acc[nt][i] = bv;
  }

#pragma unroll
  for (int k0 = 0; k0 < 64; k0 += 4) {
    const int c0 = k0 + khalf * 2;
    v2f xl = *(const v2f*)(lrow + k0);
    v2f xr = *(const v2f*)(rrow + k0);
    v2f wv = *(const v2f*)(We + c0);
    v2f av = *(const v2f*)(a1 + c0);
    v2f bv = *(const v2f*)(b1 + c0);
    v2f a;
    a.x = fmaxf((ev * wv.x + xl.x + xr.x) * av.x + bv.x, 0.0f);
    a.y = fmaxf((ev * wv.y + xl.y + xr.y) * av.y + bv.y, 0.0f);
#pragma unroll
    for (int nt = 0; nt < 4; ++nt) {
      v2f b = *(const v2f*)(Wf + (size_t)(nt * 16 + mlane) * 64 + c0);
      acc[nt] = wmma_f32(a, b, acc[nt]);
    }
  }

#pragma unroll
  for (int nt = 0; nt < 4; ++nt)
#pragma unroll
    for (int i = 0; i < 8; ++i) {
      int rr = sri[i + khalf * 8];
      atomicAdd(&conv[(size_t)rr * 64 + nt * 16 + mlane], acc[nt][i]);
    }
}

// ---------------------------------------------------------------------------
// Output MLP: h = relu([BN2(conv), right] @ W1^T + b1) ; out = relu(h @ W2^T + b2)
__global__ __launch_bounds__(32) void output_kernel(
    const float* __restrict__ conv, const float* __restrict__ a2,
    const float* __restrict__ b2, const float* __restrict__ rightf,
    const float* __restrict__ W1, const float* __restrict__ bo1,
    const float* __restrict__ W2, const float* __restrict__ bo2,
    float* __restrict__ out) {
  const int lane = threadIdx.x;
  const int mlane = lane & 15;
  const int khalf = lane >> 4;
  const int m0 = blockIdx.x * 16;

  __shared__ float hbuf[16 * 64];

  v8f acc[4];
#pragma unroll
  for (int nt = 0; nt < 4; ++nt) {
    float bv = bo1[nt * 16 + mlane];
#pragma unroll
    for (int i = 0; i < 8; ++i) acc[nt][i] = bv;
  }

  const float* crow = conv + (size_t)(m0 + mlane) * 64 + khalf * 2;
  const float* rrow = rightf + (size_t)(m0 + mlane) * 64 + khalf * 2;

  // K = 0..63 : BN2(conv) half of the concat
#pragma unroll
  for (int k0 = 0; k0 < 64; k0 += 4) {
    const int c0 = k0 + khalf * 2;
    v2f a = *(const v2f*)(crow + k0);
    v2f av = *(const v2f*)(a2 + c0);
    v2f bv = *(const v2f*)(b2 + c0);
    a.x = a.x * av.x + bv.x;
    a.y = a.y * av.y + bv.y;
#pragma unroll
    for (int nt = 0; nt < 4; ++nt) {
      v2f b = *(const v2f*)(W1 + (size_t)(nt * 16 + mlane) * 128 + c0);
      acc[nt] = wmma_f32(a, b, acc[nt]);
    }
  }
  // K = 64..127 : right_features half of the concat
#pragma unroll
  for (int k0 = 0; k0 < 64; k0 += 4) {
    v2f a = *(const v2f*)(rrow + k0);
#pragma unroll
    for (int nt = 0; nt < 4; ++nt) {
      v2f b = *(const v2f*)(W1 + (size_t)(nt * 16 + mlane) * 128 + 64 + k0 +
                            khalf * 2);
      acc[nt] = wmma_f32(a, b, acc[nt]);
    }
  }

  // relu + D-layout -> A-layout transpose through LDS
#pragma unroll
  for (int nt = 0; nt < 4; ++nt)
#pragma unroll
    for (int i = 0; i < 8; ++i)
      hbuf[(i + khalf * 8) * 64 + nt * 16 + mlane] = fmaxf(acc[nt][i], 0.0f);
  __syncthreads();

  v8f acc2[4];
#pragma unroll
  for (int nt = 0; nt < 4; ++nt) {
    float bv = bo2[nt * 16 + mlane];
#pragma unroll
    for (int i = 0; i < 8; ++i) acc2[nt][i] = bv;
  }

#pragma unroll
  for (int k0 = 0; k0 < 64; k0 += 4) {
    v2f a = *(const v2f*)(&hbuf[mlane * 64 + k0 + khalf * 2]);
#pragma unroll
    for (int nt = 0; nt < 4; ++nt) {
      v2f b = *(const v2f*)(W2 + (size_t)(nt * 16 + mlane) * 64 + k0 + khalf * 2);
      acc2[nt] = wmma_f32(a, b, acc2[nt]);
    }
  }

#pragma unroll
  for (int nt = 0; nt < 4; ++nt)
#pragma unroll
    for (int i = 0; i < 8; ++i)
      out[(size_t)(m0 + i + khalf * 8) * 64 + nt * 16 + mlane] =
          fmaxf(acc2[nt][i], 0.0f);
}

// ---------------------------------------------------------------------------
extern "C" void kernel_launch(void* const* d_in, const int* in_sizes, int n_in,
                              void* d_out, int out_size, void* d_ws,
                              size_t ws_size, hipStream_t stream) {
  (void)in_sizes; (void)n_in; (void)out_size; (void)ws_size;

  const float* leftf  = (const float*)d_in[0];
  const int*   eidx   = (const int*)d_in[1];   // [2, NE] stacked: left then right
  const float* ef     = (const float*)d_in[2];
  const float* rightf = (const float*)d_in[3];
  // d_in[4] = output_size scalar (compile-time constant NR here)
  const float* Wl  = (const float*)d_in[5];
  const float* bl  = (const float*)d_in[6];
  const float* We  = (const float*)d_in[7];
  const float* Wr  = (const float*)d_in[8];
  const float* g1  = (const float*)d_in[9];
  const float* be1 = (const float*)d_in[10];
  const float* Wf  = (const float*)d_in[11];
  const float* bfi = (const float*)d_in[12];
  const float* g2  = (const float*)d_in[13];
  const float* be2 = (const float*)d_in[14];
  const float* W1  = (const float*)d_in[15];
  const float* bo1 = (const float*)d_in[16];
  const float* W2  = (const float*)d_in[17];
  const float* bo2 = (const float*)d_in[18];
  float* out = (float*)d_out;

  // Workspace layout (floats): lp | rp | conv | stats(512)
  float* lp    = (float*)d_ws;
  float* rp    = lp + (size_t)NL * 64;
  float* conv  = rp + (size_t)NR * 64;
  float* stats = conv + (size_t)NR * 64;
  float* sum1 = stats +   0, *sq1 = stats +  64, *a1 = stats + 128, *b1 = stats + 192;
  float* sum2 = stats + 256, *sq2 = stats + 320, *a2 = stats + 384, *b2 = stats + 448;

  const int* li = eidx;
  const int* ri = eidx + NE;

  // K0: zero conv accumulator + all BN stats scratch (contiguous)
  zero_f32<<<4096, 256, 0, stream>>>(conv, NR * 64 + 512);

  // K1: projections (WMMA)
  proj16<<<NL / 16, 32, 0, stream>>>(leftf, Wl, bl, lp);
  proj16<<<NR / 16, 32, 0, stream>>>(rightf, Wr, nullptr, rp);

  // K2/K3: BN1 statistics over recomputed joint, then affine
  edge_stats<<<2048, 256, 0, stream>>>(li, ri, ef, lp, rp, We, sum1, sq1, NE);
  bn_finalize<<<1, 64, 0, stream>>>(sum1, sq1, g1, be1, (float)NE, a1, b1);

  // K4: fused BN1+relu -> W_final GEMM (WMMA) -> scatter segment_sum
  edge_gemm_scatter<<<NE / 16, 32, 0, stream>>>(li, ri, ef, lp, rp, We, a1, b1,
                                                Wf, bfi, conv);

  // K5/K6: BN2 statistics + affine
  colstats<<<1024, 256, 0, stream>>>(conv, NR, sum2, sq2);
  bn_finalize<<<1, 64, 0, stream>>>(sum2, sq2, g2, be2, (float)NR, a2, b2);

  // K7: output MLP (two chained WMMA GEMMs)
  output_kernel<<<NR / 16, 32, 0, stream>>>(conv, a2, b2, rightf, W1, bo1, W2,
                                            bo2, out);
}